// PointNetSetAbstraction_13503377178958
// MI455X (gfx1250) — compile-verified
//
#include <hip/hip_runtime.h>

// ---------------- problem constants (from reference) ----------------
#define BATCH   4
#define NPTS    8192          // N
#define SCENT   1024          // NPOINT (S)
#define KNBR    32            // NSAMPLE (K)
#define DFEAT   64            // input feature channels
#define CIN0    67            // 3 + DFEAT
#define KPAD0   68            // padded K for layer-0 GEMM
#define RADIUS2 0.01f
#define BN_EPS  1e-5f
#define MROWS   (BATCH * SCENT * KNBR)   // 131072 GEMM rows

typedef __attribute__((ext_vector_type(2))) float v2f;
typedef __attribute__((ext_vector_type(8))) float v8f;

// =====================================================================
// 1) Farthest point sampling: one block (1024 thr) per batch.
//    Writes new_xyz transposed (B,3,S) directly into d_out[0 .. 3*B*S).
// =====================================================================
__global__ void __launch_bounds__(1024)
fps_kernel(const float* __restrict__ xyz, float* __restrict__ out_newxyz) {
    const int b = blockIdx.x;
    const int t = threadIdx.x;
    const float* xb = xyz + (size_t)b * 3 * NPTS;

    // blocked ownership: thread t owns points [t*8, t*8+8) -> lowest-index
    // tie-break maps to lowest thread id.
    float px[8], py[8], pz[8], dmin[8];
#pragma unroll
    for (int i = 0; i < 8; ++i) {
        int n = t * 8 + i;
        px[i] = xb[n];
        py[i] = xb[NPTS + n];
        pz[i] = xb[2 * NPTS + n];
        dmin[i] = 1e10f;
    }

    __shared__ int   s_far;
    __shared__ float s_val[32];
    __shared__ int   s_idx[32];
    if (t == 0) s_far = 0;
    __syncthreads();

    for (int it = 0; it < SCENT; ++it) {
        const int far = s_far;
        const float cx = xb[far];
        const float cy = xb[NPTS + far];
        const float cz = xb[2 * NPTS + far];
        if (t == 0) {
            out_newxyz[(b * 3 + 0) * SCENT + it] = cx;
            out_newxyz[(b * 3 + 1) * SCENT + it] = cy;
            out_newxyz[(b * 3 + 2) * SCENT + it] = cz;
        }
        float best = -1.0f;
        int   bidx = 0;
#pragma unroll
        for (int i = 0; i < 8; ++i) {
            float dx = px[i] - cx, dy = py[i] - cy, dz = pz[i] - cz;
            float d = dx * dx + dy * dy + dz * dz;
            dmin[i] = fminf(dmin[i], d);
            if (dmin[i] > best) { best = dmin[i]; bidx = t * 8 + i; }
        }
        // wave32 argmax (prefer smaller index on ties -> matches jnp.argmax)
#pragma unroll
        for (int off = 16; off > 0; off >>= 1) {
            float ov = __shfl_down(best, off, 32);
            int   oi = __shfl_down(bidx, off, 32);
            if (ov > best || (ov == best && oi < bidx)) { best = ov; bidx = oi; }
        }
        const int wv = t >> 5;
        if ((t & 31) == 0) { s_val[wv] = best; s_idx[wv] = bidx; }
        __syncthreads();
        if (t < 32) {
            best = s_val[t];
            bidx = s_idx[t];
#pragma unroll
            for (int off = 16; off > 0; off >>= 1) {
                float ov = __shfl_down(best, off, 32);
                int   oi = __shfl_down(bidx, off, 32);
                if (ov > best || (ov == best && oi < bidx)) { best = ov; bidx = oi; }
            }
            if (t == 0) s_far = bidx;
        }
        __syncthreads();
    }
}

// =====================================================================
// 2) Ball query + grouping: one wave per centroid. Collect first KNBR
//    in-radius indices in ascending order (ballot + prefix popcount),
//    pad with the first hit, gather [xyz-c, feats] into h (row-major,
//    KPAD0 stride, channel 67 zero-padded).
// =====================================================================
__global__ void __launch_bounds__(256)
ballquery_group_kernel(const float* __restrict__ xyz,
                       const float* __restrict__ points,
                       const float* __restrict__ newxyz,   // (B,3,S) in d_out
                       float* __restrict__ h) {
    const int gid  = blockIdx.x * 8 + (threadIdx.x >> 5);  // centroid id
    const int lane = threadIdx.x & 31;
    const int wv   = threadIdx.x >> 5;
    if (gid >= BATCH * SCENT) return;
    const int b = gid / SCENT;
    const int s = gid % SCENT;

    const float* xb = xyz + (size_t)b * 3 * NPTS;
    const float cx = newxyz[(b * 3 + 0) * SCENT + s];
    const float cy = newxyz[(b * 3 + 1) * SCENT + s];
    const float cz = newxyz[(b * 3 + 2) * SCENT + s];

    __shared__ int s_gi[8][KNBR];

    int count = 0;
    for (int n0 = 0; n0 < NPTS && count < KNBR; n0 += 32) {
        const int n = n0 + lane;
        float dx = xb[n] - cx, dy = xb[NPTS + n] - cy, dz = xb[2 * NPTS + n] - cz;
        bool in = (dx * dx + dy * dy + dz * dz) <= RADIUS2;
        unsigned mask = (unsigned)__ballot(in);
        if (in) {
            int pos = count + __popc(mask & ((1u << lane) - 1u));
            if (pos < KNBR) s_gi[wv][pos] = n;
        }
        count += __popc(mask);
    }
    // in-wave LDS scatter -> make lanes' stores visible before reading
    asm volatile("s_wait_dscnt 0" ::: "memory");

    const int cnt  = count < KNBR ? count : KNBR;
    const int gi_k = s_gi[wv][lane < cnt ? lane : 0];

    const size_t row = (size_t)gid * KNBR + lane;
    float* hr = h + row * KPAD0;
    hr[0] = xb[gi_k] - cx;
    hr[1] = xb[NPTS + gi_k] - cy;
    hr[2] = xb[2 * NPTS + gi_k] - cz;
    const float* pb = points + (size_t)b * DFEAT * NPTS;
#pragma unroll 4
    for (int c = 0; c < DFEAT; ++c) hr[3 + c] = pb[(size_t)c * NPTS + gi_k];
    hr[CIN0] = 0.0f;   // pad channel
}

// =====================================================================
// 3) GEMM via V_WMMA_F32_16X16X4_F32:  Y[M,N] = A[M,K] * W^T + bias
//    One wave per 16-row M-tile; accumulates all NDIM/16 N-tiles.
// =====================================================================
template <int NDIM, int KREAL, int KPAD>
__global__ void __launch_bounds__(256)
gemm_wmma_kernel(const float* __restrict__ A, int lda,
                 const float* __restrict__ W,      // (NDIM, KREAL) row-major
                 const float* __restrict__ bias,   // (NDIM)
                 float* __restrict__ Y, int ldy) {
    constexpr int NT = NDIM / 16;
    const int wave  = threadIdx.x >> 5;
    const int lane  = threadIdx.x & 31;
    const int mbase = (blockIdx.x * 8 + wave) * 16;
    const int nlo   = lane & 15;     // A row / D column within tile
    const int hi    = lane >> 4;     // K-half select

    v8f acc[NT];
#pragma unroll
    for (int t = 0; t < NT; ++t) {
        const float bv = bias[t * 16 + nlo];
#pragma unroll
        for (int j = 0; j < 8; ++j) acc[t][j] = bv;
    }

    const float* arow = A + (size_t)(mbase + nlo) * lda;
    for (int k0 = 0; k0 < KPAD; k0 += 4) {
        const int ka = k0 + 2 * hi;
        v2f a;
        a.x = arow[ka];
        a.y = arow[ka + 1];
#pragma unroll
        for (int t = 0; t < NT; ++t) {
            const int col = t * 16 + nlo;
            v2f bb;
            bb.x = (ka     < KREAL) ? W[(size_t)col * KREAL + ka]     : 0.0f;
            bb.y = (ka + 1 < KREAL) ? W[(size_t)col * KREAL + ka + 1] : 0.0f;
            acc[t] = __builtin_amdgcn_wmma_f32_16x16x4_f32(
                false, a, false, bb, (short)0, acc[t], false, false);
        }
    }

#pragma unroll
    for (int t = 0; t < NT; ++t)
#pragma unroll
        for (int j = 0; j < 8; ++j)
            Y[(size_t)(mbase + j + 8 * hi) * ldy + t * 16 + nlo] = acc[t][j];
}

// =====================================================================
// 4) Per-channel sum / sum-of-squares (batch-norm training stats)
// =====================================================================
__global__ void __launch_bounds__(256)
zero_stats_kernel(float* stats, int n) {
    int i = blockIdx.x * 256 + threadIdx.x;
    if (i < n) stats[i] = 0.0f;
}

__global__ void __launch_bounds__(256)
stats_kernel(const float* __restrict__ Y, int ldy,
             float* __restrict__ sum, float* __restrict__ sumsq) {
    const int c  = blockIdx.x;
    const int t  = threadIdx.x;
    const int r0 = blockIdx.y * 2048;
    float s = 0.0f, s2 = 0.0f;
    for (int r = r0 + t; r < r0 + 2048; r += 256) {
        float v = Y[(size_t)r * ldy + c];
        s += v;
        s2 += v * v;
    }
    __shared__ float ls[256], ls2[256];
    ls[t] = s; ls2[t] = s2;
    __syncthreads();
    for (int off = 128; off > 0; off >>= 1) {
        if (t < off) { ls[t] += ls[t + off]; ls2[t] += ls2[t + off]; }
        __syncthreads();
    }
    if (t == 0) {
        atomicAdd(&sum[c],   ls[0]);
        atomicAdd(&sumsq[c], ls2[0]);
    }
}

// =====================================================================
// 5) BatchNorm (training stats) + ReLU, in place
// =====================================================================
__global__ void __launch_bounds__(256)
bn_relu_kernel(float* __restrict__ Y, int Cdim,
               const float* __restrict__ sum, const float* __restrict__ sumsq,
               const float* __restrict__ g, const float* __restrict__ beta) {
    const size_t i = (size_t)blockIdx.x * 256 + threadIdx.x;
    if (i >= (size_t)MROWS * Cdim) return;
    const int c = (int)(i % Cdim);
    const float inv = 1.0f / (float)MROWS;
    const float mean = sum[c] * inv;
    const float var  = sumsq[c] * inv - mean * mean;
    const float v = Y[i];
    const float o = (v - mean) * rsqrtf(var + BN_EPS) * g[c] + beta[c];
    Y[i] = o > 0.0f ? o : 0.0f;
}

// =====================================================================
// 6) Max over K neighbors + transpose -> d_out part 2, (B,128,S)
// =====================================================================
__global__ void __launch_bounds__(256)
maxpool_kernel(const float* __restrict__ Y2, float* __restrict__ out) {
    const int i = blockIdx.x * 256 + threadIdx.x;   // over B*S*128
    if (i >= BATCH * SCENT * 128) return;
    const int c  = i & 127;
    const int bs = i >> 7;
    const int b  = bs / SCENT;
    const int s  = bs % SCENT;
    const float* p = Y2 + (size_t)bs * KNBR * 128 + c;
    float m = p[0];
#pragma unroll 8
    for (int k = 1; k < KNBR; ++k) m = fmaxf(m, p[(size_t)k * 128]);
    out[((size_t)b * 128 + c) * SCENT + s] = m;
}

// =====================================================================
// Host-side launch sequence
// =====================================================================
extern "C" void kernel_launch(void* const* d_in, const int* in_sizes, int n_in,
                              void* d_out, int out_size, void* d_ws, size_t ws_size,
                              hipStream_t stream) {
    const float* xyz    = (const float*)d_in[0];
    const float* points = (const float*)d_in[1];
    const float* w0 = (const float*)d_in[2];
    const float* b0 = (const float*)d_in[3];
    const float* g0 = (const float*)d_in[4];
    const float* be0 = (const float*)d_in[5];
    const float* w1 = (const float*)d_in[6];
    const float* b1 = (const float*)d_in[7];
    const float* g1 = (const float*)d_in[8];
    const float* be1 = (const float*)d_in[9];
    const float* w2 = (const float*)d_in[10];
    const float* b2 = (const float*)d_in[11];
    const float* g2 = (const float*)d_in[12];
    const float* be2 = (const float*)d_in[13];

    float* out_newxyz = (float*)d_out;                              // B*3*S
    float* out_feat   = (float*)d_out + (size_t)BATCH * 3 * SCENT;  // B*128*S

    // workspace layout
    char* ws = (char*)d_ws;
    float* hbuf = (float*)ws;                                        // M x 68
    const size_t H_BYTES = (size_t)MROWS * KPAD0 * sizeof(float);
    float* ybuf = (float*)(ws + H_BYTES);                            // M x 128
    const size_t Y_BYTES = (size_t)MROWS * 128 * sizeof(float);
    float* stats = (float*)(ws + H_BYTES + Y_BYTES);                 // 512 floats
    float* sum0 = stats,        *sq0 = stats + 64;
    float* sum1 = stats + 128,  *sq1 = stats + 192;
    float* sum2 = stats + 256,  *sq2 = stats + 384;

    // 1) FPS -> new_xyz (also output part 1)
    fps_kernel<<<BATCH, 1024, 0, stream>>>(xyz, out_newxyz);

    // 2) ball query + grouping -> h (M x 68)
    ballquery_group_kernel<<<(BATCH * SCENT) / 8, 256, 0, stream>>>(
        xyz, points, out_newxyz, hbuf);

    // zero BN statistics
    zero_stats_kernel<<<2, 256, 0, stream>>>(stats, 512);

    const int gemm_blocks = MROWS / (8 * 16);   // 1024

    // layer 0: h(M,68) @ w0^T -> y0 (M,64) in ybuf
    gemm_wmma_kernel<64, CIN0, KPAD0><<<gemm_blocks, 256, 0, stream>>>(
        hbuf, KPAD0, w0, b0, ybuf, 64);
    stats_kernel<<<dim3(64, MROWS / 2048), 256, 0, stream>>>(ybuf, 64, sum0, sq0);
    bn_relu_kernel<<<(MROWS * 64) / 256, 256, 0, stream>>>(ybuf, 64, sum0, sq0, g0, be0);

    // layer 1: y0(M,64) @ w1^T -> y1 (M,64) reusing h region
    gemm_wmma_kernel<64, 64, 64><<<gemm_blocks, 256, 0, stream>>>(
        ybuf, 64, w1, b1, hbuf, 64);
    stats_kernel<<<dim3(64, MROWS / 2048), 256, 0, stream>>>(hbuf, 64, sum1, sq1);
    bn_relu_kernel<<<(MROWS * 64) / 256, 256, 0, stream>>>(hbuf, 64, sum1, sq1, g1, be1);

    // layer 2: y1(M,64) @ w2^T -> y2 (M,128) in ybuf
    gemm_wmma_kernel<128, 64, 64><<<gemm_blocks, 256, 0, stream>>>(
        hbuf, 64, w2, b2, ybuf, 128);
    stats_kernel<<<dim3(128, MROWS / 2048), 256, 0, stream>>>(ybuf, 128, sum2, sq2);
    bn_relu_kernel<<<(MROWS * 128) / 256, 256, 0, stream>>>(ybuf, 128, sum2, sq2, g2, be2);

    // max over K + transpose -> output part 2
    maxpool_kernel<<<(BATCH * SCENT * 128) / 256, 256, 0, stream>>>(ybuf, out_feat);
}